// StaticDecEmbedding_67259187855533
// MI455X (gfx1250) — compile-verified
//
#include <hip/hip_runtime.h>

// Problem sizes from the reference
#define B_ 8192
#define F_ 32
#define E_ 256
#define H_ 64

typedef __attribute__((ext_vector_type(2))) float v2f;
typedef __attribute__((ext_vector_type(4))) float v4f;
typedef __attribute__((ext_vector_type(8))) float v8f;

// ---------------------------------------------------------------------------
// Kernel 1: collapse the two linear layers.
//   v[f,e] = sum_h W1[f,e,h] * W2[f,h]
//   c[f]   = sum_h b1[f,h]  * W2[f,h] + b2[f]
// 32 blocks x 256 threads; reads 2 MiB of W1 once. Negligible cost.
// ---------------------------------------------------------------------------
__global__ __launch_bounds__(256) void precompute_vw(
    const float* __restrict__ W1, const float* __restrict__ b1,
    const float* __restrict__ W2, const float* __restrict__ b2,
    float* __restrict__ v, float* __restrict__ cb) {
  const int f = blockIdx.x;
  const int e = threadIdx.x;
  const float* w1 = W1 + ((size_t)f * E_ + e) * H_;
  const float* w2 = W2 + (size_t)f * H_;
  float s = 0.f;
#pragma unroll 8
  for (int h = 0; h < H_; ++h) s = fmaf(w1[h], w2[h], s);
  v[(size_t)f * E_ + e] = s;
  if (e == 0) {
    float t = b2[f];
    const float* bb = b1 + (size_t)f * H_;
#pragma unroll 8
    for (int h = 0; h < H_; ++h) t = fmaf(bb[h], w2[h], t);
    cb[f] = t;
  }
}

// ---------------------------------------------------------------------------
// Kernel 2: out[b,f] = dot(x[b,f,:], v[f,:]) + c[f] via chained
// V_WMMA_F32_16X16X4_F32. One wave = 16 batch rows x 1 feature.
// A[m,k] = x[b_m, f, e(k)], B[k,n] = v[f, e(k)] (broadcast over n).
// Lane L loads a float4 at e = 8j + 4*(L/16), feeding two WMMAs per j:
//   WMMA0 uses (.x,.y)  -> e(k=0..3) = {8j, 8j+1, 8j+4, 8j+5}
//   WMMA1 uses (.z,.w)  -> e(k=0..3) = {8j+2, 8j+3, 8j+6, 8j+7}
// B uses the identical e-permutation, so the contraction covers e=[8j,8j+8).
// ---------------------------------------------------------------------------
__global__ __launch_bounds__(128) void fused_mlp_gemv(
    const float* __restrict__ x, const float* __restrict__ v,
    const float* __restrict__ cb, float* __restrict__ out) {
  const int f    = blockIdx.y;
  const int wave = threadIdx.x >> 5;
  const int lane = threadIdx.x & 31;
  const int half = lane >> 4;   // 0: lanes 0-15 (K=0,1), 1: lanes 16-31 (K=2,3)
  const int m    = lane & 15;   // A-matrix row (batch index within tile)
  const int b0   = (blockIdx.x * 4 + wave) * 16;

  // Stage v[f][0:256] (1 KB) into LDS; shared by all 4 waves of the block.
  __shared__ float sv[E_];
#pragma unroll
  for (int i = threadIdx.x; i < E_; i += 128) sv[i] = v[(size_t)f * E_ + i];
  __syncthreads();

  const float* xrow = x + (((size_t)(b0 + m)) * F_ + f) * E_;

  v8f acc = {};
#pragma unroll
  for (int j = 0; j < E_ / 8; ++j) {
    const int e = 8 * j + 4 * half;
    // x is 256 MiB streamed exactly once (> 192 MB L2): non-temporal load.
    v4f xa = __builtin_nontemporal_load((const v4f*)(xrow + e));
    v4f vb = *(const v4f*)(&sv[e]);  // ds_load_b128, half-wave broadcast
    v2f a0 = {xa.x, xa.y};
    v2f a1 = {xa.z, xa.w};
    v2f q0 = {vb.x, vb.y};
    v2f q1 = {vb.z, vb.w};
    acc = __builtin_amdgcn_wmma_f32_16x16x4_f32(
        /*neg_a=*/false, a0, /*neg_b=*/false, q0,
        /*c_mod=*/(short)0, acc, /*reuse_a=*/false, /*reuse_b=*/false);
    acc = __builtin_amdgcn_wmma_f32_16x16x4_f32(
        false, a1, false, q1, (short)0, acc, false, false);
  }

  // All 16 D columns are identical. C layout: VGPR r at lane L holds
  // D[M = r + 8*(L/16), N = L%16]. Pick row (lane&7)+8*half via an unrolled
  // select chain; lanes {0..7, 16..23} each write one output.
  float val = acc[0];
  if ((lane & 7) == 1) val = acc[1];
  if ((lane & 7) == 2) val = acc[2];
  if ((lane & 7) == 3) val = acc[3];
  if ((lane & 7) == 4) val = acc[4];
  if ((lane & 7) == 5) val = acc[5];
  if ((lane & 7) == 6) val = acc[6];
  if ((lane & 7) == 7) val = acc[7];

  if ((lane & 15) < 8) {
    const int mm = (lane & 7) + 8 * half;
    out[(size_t)(b0 + mm) * F_ + f] = val + cb[f];
  }
}

// ---------------------------------------------------------------------------
extern "C" void kernel_launch(void* const* d_in, const int* in_sizes, int n_in,
                              void* d_out, int out_size, void* d_ws,
                              size_t ws_size, hipStream_t stream) {
  const float* x  = (const float*)d_in[0];
  const float* W1 = (const float*)d_in[1];
  const float* b1 = (const float*)d_in[2];
  const float* W2 = (const float*)d_in[3];
  const float* b2 = (const float*)d_in[4];
  float* out = (float*)d_out;

  float* v  = (float*)d_ws;        // F_*E_ floats = 32 KB
  float* cb = v + (size_t)F_ * E_; // F_ floats

  precompute_vw<<<F_, E_, 0, stream>>>(W1, b1, W2, b2, v, cb);

  dim3 grid(B_ / (16 * 4), F_);    // 4 waves/block, 16 b-rows/wave
  fused_mlp_gemv<<<grid, 128, 0, stream>>>(x, v, cb, out);
}